// CSNetModel_3418793968216
// MI455X (gfx1250) — compile-verified
//
#include <hip/hip_runtime.h>
#include <hip/hip_bf16.h>
#include <math.h>

#define N_NODES 100000
#define DFEAT   128
#define E_HOMO  600000
#define E_REL_N 150000
#define R_HET   4
#define R_RG    8

typedef __attribute__((ext_vector_type(2))) float v2f;
typedef __attribute__((ext_vector_type(8))) float v8f;

// ---------------------------------------------------------------------------
// zero fill (float4 stores)
// ---------------------------------------------------------------------------
__global__ __launch_bounds__(256) void zero_kernel(float* __restrict__ p, int n4) {
    int i = blockIdx.x * blockDim.x + threadIdx.x;
    if (i < n4) {
        float4 z; z.x = 0.f; z.y = 0.f; z.z = 0.f; z.w = 0.f;
        ((float4*)p)[i] = z;
    }
}

// ---------------------------------------------------------------------------
// segment_sum scatter: 1 warp (wave32) per edge, lane covers 4 consecutive
// features -> float4 gather of src row + 4x global_atomic_add_f32 to dst row.
// Optional etype filter (et != nullptr) selects one relation per pass.
// ---------------------------------------------------------------------------
__global__ __launch_bounds__(256) void scatter_add_kernel(
    const float* __restrict__ x, const int* __restrict__ src,
    const int* __restrict__ dst, const int* __restrict__ et,
    int rel, int nE, float* __restrict__ agg) {
    int gid = blockIdx.x * blockDim.x + threadIdx.x;
    int e = gid >> 5;
    if (e >= nE) return;
    if (et && et[e] != rel) return;
    int lane = gid & 31;
    int c = lane * 4;
    int s = src[e];
    int d = dst[e];
    const float4 v = *(const float4*)(x + (size_t)s * DFEAT + c);
    float* o = agg + (size_t)d * DFEAT + c;
    atomicAdd(o + 0, v.x);
    atomicAdd(o + 1, v.y);
    atomicAdd(o + 2, v.z);
    atomicAdd(o + 3, v.w);
}

// ---------------------------------------------------------------------------
// Cout = [accum?] out_scale * f( (Cin? Cin:0) + A[N,128] @ W[128,128] + bias )
// f = tanh if do_tanh.  WMMA f32 16x16x4 path.
// Block = 256 threads = 8 waves; block -> one 16-row M tile, wave w -> N tile w.
// Fragment layout per ISA 7.12.2:
//   A 16x4 : lanes 0-15 {K=k+0,k+1}, lanes 16-31 {K=k+2,k+3}, row = m0+lane%16
//   B 4x16 : mirrored (col = n0+lane%16)
//   C/D    : elem v -> row v (lanes 0-15) / v+8 (lanes 16-31), col = lane%16
// ---------------------------------------------------------------------------
__global__ __launch_bounds__(256) void gemm128_wmma_kernel(
    const float* __restrict__ A, const float* __restrict__ W,
    const float* __restrict__ bias, const float* __restrict__ Cin,
    float* __restrict__ Cout, int do_tanh, float out_scale, int out_accum) {
    const int lane = threadIdx.x & 31;
    const int wave = threadIdx.x >> 5;        // 0..7 -> N tile
    const int m0   = blockIdx.x * 16;
    const int n0   = wave * 16;
    const int half = (lane < 16) ? 0 : 1;
    const int l15  = lane & 15;
    const int koff = half * 2;

    v8f acc = {};
    if (Cin) {
#pragma unroll
        for (int v = 0; v < 8; ++v) {
            int row = m0 + v + half * 8;
            acc[v] = Cin[(size_t)row * DFEAT + n0 + l15];
        }
    }

    const float* arow = A + (size_t)(m0 + l15) * DFEAT;
#pragma unroll 4
    for (int k = 0; k < DFEAT; k += 4) {
        v2f a, b;
        const float* ap = arow + k + koff;
        a.x = ap[0];
        a.y = ap[1];
        const float* bp = W + (size_t)(k + koff) * DFEAT + n0 + l15;
        b.x = bp[0];
        b.y = bp[DFEAT];
        acc = __builtin_amdgcn_wmma_f32_16x16x4_f32(
            false, a, false, b, (short)0, acc, false, false);
    }

    float bval = bias ? bias[n0 + l15] : 0.0f;
#pragma unroll
    for (int v = 0; v < 8; ++v) {
        int row = m0 + v + half * 8;
        float r = acc[v] + bval;
        if (do_tanh) r = tanhf(r);
        r *= out_scale;
        float* o = Cout + (size_t)row * DFEAT + n0 + l15;
        if (out_accum) *o += r; else *o = r;
    }
}

// ---------------------------------------------------------------------------
extern "C" void kernel_launch(void* const* d_in, const int* in_sizes, int n_in,
                              void* d_out, int out_size, void* d_ws, size_t ws_size,
                              hipStream_t stream) {
    // inputs (setup_inputs order)
    const int* gcn_src1 = (const int*)d_in[0];
    const int* gcn_dst1 = (const int*)d_in[1];
    const int* gcn_src2 = (const int*)d_in[2];
    const int* gcn_dst2 = (const int*)d_in[3];
    const int* rg_src1  = (const int*)d_in[4];
    const int* rg_dst1  = (const int*)d_in[5];
    const int* rg_et1   = (const int*)d_in[6];
    const int* rg_src2  = (const int*)d_in[7];
    const int* rg_dst2  = (const int*)d_in[8];
    const int* rg_et2   = (const int*)d_in[9];
    const int* het_src1 = (const int*)d_in[10];
    const int* het_dst1 = (const int*)d_in[11];
    const int* het_src2 = (const int*)d_in[12];
    const int* het_dst2 = (const int*)d_in[13];
    const float* emb      = (const float*)d_in[14];
    const float* gcn_W1   = (const float*)d_in[15];
    const float* gcn_b1   = (const float*)d_in[16];
    const float* gcn_W2   = (const float*)d_in[17];
    const float* gcn_b2   = (const float*)d_in[18];
    const float* rg_W1    = (const float*)d_in[19];
    const float* rg_loop1 = (const float*)d_in[20];
    const float* rg_b1    = (const float*)d_in[21];
    const float* rg_W2    = (const float*)d_in[22];
    const float* rg_loop2 = (const float*)d_in[23];
    const float* rg_b2    = (const float*)d_in[24];
    const float* het_W1   = (const float*)d_in[25];
    const float* het_b1   = (const float*)d_in[26];
    const float* het_W2   = (const float*)d_in[27];
    const float* het_b2   = (const float*)d_in[28];

    const size_t NM = (size_t)N_NODES * DFEAT;   // 12.8M floats
    float* out = (float*)d_out;
    float* hcf = out;
    float* hc  = out + NM;
    float* hs  = out + 2 * NM;

    float* ws   = (float*)d_ws;
    float* agg  = ws;            // [N,128] scatter accumulator
    float* pre  = ws + NM;       // [N,128] pre-activation accumulator
    float* hbuf = ws + 2 * NM;   // [N,128] hidden activations

    const dim3 blk(256);
    const int zeroBlocks  = (int)((NM / 4 + 255) / 256);
    const int scatBlocksE = (E_HOMO * 32 + 255) / 256;
    const int scatBlocksR = (E_REL_N * 32 + 255) / 256;
    const int gemmBlocks  = N_NODES / 16;        // 6250 (exact)

#define ZERO(p) zero_kernel<<<zeroBlocks, blk, 0, stream>>>((p), (int)(NM / 4))
#define SCAT(x, s, d, et, r, ne, nb) \
    scatter_add_kernel<<<(nb), blk, 0, stream>>>((x), (s), (d), (et), (r), (ne), agg)
#define GEMM(A, W, B, CI, CO, T, S, AC) \
    gemm128_wmma_kernel<<<gemmBlocks, blk, 0, stream>>>((A), (W), (B), (CI), (CO), (T), (S), (AC))

    // ---------------- GCN branch -> hcf ----------------
    ZERO(agg);
    SCAT(emb, gcn_src1, gcn_dst1, (const int*)nullptr, 0, E_HOMO, scatBlocksE);
    GEMM(agg, gcn_W1, gcn_b1, (const float*)nullptr, hbuf, 1, 1.0f, 0);
    ZERO(agg);
    SCAT(hbuf, gcn_src2, gcn_dst2, (const int*)nullptr, 0, E_HOMO, scatBlocksE);
    GEMM(agg, gcn_W2, gcn_b2, (const float*)nullptr, hcf, 1, 1.0f, 0);

    // ---------------- RGCN branch -> hc ----------------
    // layer 1: pre = emb @ loop1 + b1; pre += agg_r @ W1[r]; h = tanh(pre)
    GEMM(emb, rg_loop1, rg_b1, (const float*)nullptr, pre, 0, 1.0f, 0);
    for (int r = 0; r < R_RG; ++r) {
        ZERO(agg);
        SCAT(emb, rg_src1, rg_dst1, rg_et1, r, E_HOMO, scatBlocksE);
        const float* Wr = rg_W1 + (size_t)r * DFEAT * DFEAT;
        if (r < R_RG - 1)
            GEMM(agg, Wr, (const float*)nullptr, pre, pre, 0, 1.0f, 0);
        else
            GEMM(agg, Wr, (const float*)nullptr, pre, hbuf, 1, 1.0f, 0);
    }
    // layer 2
    GEMM(hbuf, rg_loop2, rg_b2, (const float*)nullptr, pre, 0, 1.0f, 0);
    for (int r = 0; r < R_RG; ++r) {
        ZERO(agg);
        SCAT(hbuf, rg_src2, rg_dst2, rg_et2, r, E_HOMO, scatBlocksE);
        const float* Wr = rg_W2 + (size_t)r * DFEAT * DFEAT;
        if (r < R_RG - 1)
            GEMM(agg, Wr, (const float*)nullptr, pre, pre, 0, 1.0f, 0);
        else
            GEMM(agg, Wr, (const float*)nullptr, pre, hc, 1, 1.0f, 0);
    }

    // ---------------- Hetero branch -> hs ----------------
    const float invR = 1.0f / (float)R_HET;
    // layer 1: hbuf = mean_r tanh(agg_r @ W1[r] + b1[r])
    ZERO(hbuf);
    for (int r = 0; r < R_HET; ++r) {
        ZERO(agg);
        SCAT(emb, het_src1 + (size_t)r * E_REL_N, het_dst1 + (size_t)r * E_REL_N,
             (const int*)nullptr, 0, E_REL_N, scatBlocksR);
        GEMM(agg, het_W1 + (size_t)r * DFEAT * DFEAT, het_b1 + (size_t)r * DFEAT,
             (const float*)nullptr, hbuf, 1, invR, 1);
    }
    // layer 2: hs = mean_r tanh(agg_r @ W2[r] + b2[r])
    ZERO(hs);
    for (int r = 0; r < R_HET; ++r) {
        ZERO(agg);
        SCAT(hbuf, het_src2 + (size_t)r * E_REL_N, het_dst2 + (size_t)r * E_REL_N,
             (const int*)nullptr, 0, E_REL_N, scatBlocksR);
        GEMM(agg, het_W2 + (size_t)r * DFEAT * DFEAT, het_b2 + (size_t)r * DFEAT,
             (const float*)nullptr, hs, 1, invR, 1);
    }

#undef ZERO
#undef SCAT
#undef GEMM
}